// ImplicitModelLoRA_41601053229644
// MI455X (gfx1250) — compile-verified
//
#include <hip/hip_runtime.h>

typedef float v2f __attribute__((ext_vector_type(2)));
typedef float v8f __attribute__((ext_vector_type(8)));

// ---------------------------------------------------------------------------
// CDNA5 f32 WMMA: D(16x16 f32) = A(16x4 f32) * B(4x16 f32) + C
// 8-arg form: (neg_a, A, neg_b, B, c_mod, C, reuse_a, reuse_b)
// ---------------------------------------------------------------------------
__device__ __forceinline__ v8f wmma_f32(v2f a, v2f b, v8f c) {
  return __builtin_amdgcn_wmma_f32_16x16x4_f32(false, a, false, b, (short)0, c,
                                               false, false);
}

// A-fragment (16x4 tile) from row-major matrix with leading dim ld.
// Lane r=lane&15 holds row mTile+r; VGPR v holds K = k0 + 2*half + v.
__device__ __forceinline__ v2f ldfragA(const float* __restrict__ M, int row,
                                       int k0, int ld, int half) {
  const float* p = M + (size_t)row * ld + k0 + 2 * half;
  v2f v;
  v.x = p[0];
  v.y = p[1];
  return v;
}

// B-fragment (4x16 tile) from row-major K x N matrix (element [k][n]).
__device__ __forceinline__ v2f ldfragB_kn(const float* __restrict__ M, int col,
                                          int k0, int ld, int half) {
  v2f v;
  v.x = M[(size_t)(k0 + 2 * half) * ld + col];
  v.y = M[(size_t)(k0 + 2 * half + 1) * ld + col];
  return v;
}

// B-fragment (4x16 tile) from row-major N x K matrix (implicit transpose).
__device__ __forceinline__ v2f ldfragB_nk(const float* __restrict__ M, int col,
                                          int k0, int ld, int half) {
  const float* p = M + (size_t)col * ld + k0 + 2 * half;
  v2f v;
  v.x = p[0];
  v.y = p[1];
  return v;
}

// ---------------------------------------------------------------------------
// Kernel 1: L-inf matrix norms (max abs row-sum).
// block 0: norm of L (1024x64)  -> scal[0]
// block 1: norm of R^T (64x1024): column sums of R -> scal[1]
// ---------------------------------------------------------------------------
__global__ void __launch_bounds__(1024) norms_kernel(
    const float* __restrict__ L, const float* __restrict__ R,
    float* __restrict__ scal) {
  __shared__ float red[1024];
  const int t = threadIdx.x;
  float s = 0.0f;
  if (blockIdx.x == 0) {
    const float* row = L + (size_t)t * 64;
    for (int k = 0; k < 64; ++k) s += fabsf(row[k]);
  } else {
    if (t < 64) {
      for (int i = 0; i < 1024; ++i) s += fabsf(R[(size_t)i * 64 + t]);
    }
  }
  red[t] = s;
  __syncthreads();
  for (int off = 512; off > 0; off >>= 1) {
    if (t < off) red[t] = fmaxf(red[t], red[t + off]);
    __syncthreads();
  }
  if (t == 0) scal[blockIdx.x] = red[0];
}

// ---------------------------------------------------------------------------
// Kernel 2: Amat = s * (L @ R^T), 1024x1024, K=64.
// One wave -> 16x64 tile (4 accumulators). 1024 waves = 128 blocks x 8 waves.
// ---------------------------------------------------------------------------
__global__ void __launch_bounds__(256) amat_kernel(
    const float* __restrict__ L, const float* __restrict__ R,
    const float* __restrict__ scal, float* __restrict__ Amat) {
  const int lane = threadIdx.x & 31;
  const int w = threadIdx.x >> 5;
  const int half = lane >> 4;
  const int r = lane & 15;
  const int waveId = blockIdx.x * 8 + w;
  const int mTile = (waveId >> 4) * 16;  // 64 M tiles
  const int nTile = (waveId & 15) * 64;  // 16 N tiles

  const float nL = scal[0], nR = scal[1];
  const float s = ((nL > 0.97f) ? (0.97f / nL) : 1.0f) *
                  ((nR > 0.97f) ? (0.97f / nR) : 1.0f);

  v8f c0 = {}, c1 = {}, c2 = {}, c3 = {};
  for (int k0 = 0; k0 < 64; k0 += 4) {
    v2f a = ldfragA(L, mTile + r, k0, 64, half);
    v2f b0 = ldfragB_nk(R, nTile + r, k0, 64, half);
    v2f b1 = ldfragB_nk(R, nTile + r + 16, k0, 64, half);
    v2f b2 = ldfragB_nk(R, nTile + r + 32, k0, 64, half);
    v2f b3 = ldfragB_nk(R, nTile + r + 48, k0, 64, half);
    c0 = wmma_f32(a, b0, c0);
    c1 = wmma_f32(a, b1, c1);
    c2 = wmma_f32(a, b2, c2);
    c3 = wmma_f32(a, b3, c3);
  }
  const int col = nTile + r;
#pragma unroll
  for (int i = 0; i < 8; ++i) {
    const int row = mTile + half * 8 + i;
    float* o = Amat + (size_t)row * 1024 + col;
    o[0] = c0[i] * s;
    o[16] = c1[i] * s;
    o[32] = c2[i] * s;
    o[48] = c3[i] * s;
  }
}

// ---------------------------------------------------------------------------
// Kernel 3: BU = B @ U^T, 1024x256, K=512. 256 waves = 32 blocks.
// ---------------------------------------------------------------------------
__global__ void __launch_bounds__(256) bu_kernel(const float* __restrict__ B,
                                                 const float* __restrict__ U,
                                                 float* __restrict__ BU) {
  const int lane = threadIdx.x & 31;
  const int w = threadIdx.x >> 5;
  const int half = lane >> 4;
  const int r = lane & 15;
  const int waveId = blockIdx.x * 8 + w;
  const int mTile = (waveId >> 2) * 16;  // 64 M tiles
  const int nTile = (waveId & 3) * 64;   // 4 N tiles

  v8f c0 = {}, c1 = {}, c2 = {}, c3 = {};
  for (int k0 = 0; k0 < 512; k0 += 4) {
    v2f a = ldfragA(B, mTile + r, k0, 512, half);
    v2f b0 = ldfragB_nk(U, nTile + r, k0, 512, half);
    v2f b1 = ldfragB_nk(U, nTile + r + 16, k0, 512, half);
    v2f b2 = ldfragB_nk(U, nTile + r + 32, k0, 512, half);
    v2f b3 = ldfragB_nk(U, nTile + r + 48, k0, 512, half);
    c0 = wmma_f32(a, b0, c0);
    c1 = wmma_f32(a, b1, c1);
    c2 = wmma_f32(a, b2, c2);
    c3 = wmma_f32(a, b3, c3);
  }
  const int col = nTile + r;
#pragma unroll
  for (int i = 0; i < 8; ++i) {
    const int row = mTile + half * 8 + i;
    float* o = BU + (size_t)row * 256 + col;
    o[0] = c0[i];
    o[16] = c1[i];
    o[32] = c2[i];
    o[48] = c3[i];
  }
}

// ---------------------------------------------------------------------------
// Kernel 4: X = relu(BU)  (first fixed-point iteration from X0 = 0)
// ---------------------------------------------------------------------------
__global__ void relu_copy_kernel(const float* __restrict__ in,
                                 float* __restrict__ out, int n) {
  int i = blockIdx.x * blockDim.x + threadIdx.x;
  if (i < n) out[i] = fmaxf(in[i], 0.0f);
}

// ---------------------------------------------------------------------------
// Kernel 5 (x149): Xout = relu(A @ Xin + BU). M=1024, N=256, K=1024.
// One wave -> 16x64 tile, 4 f32 WMMA accumulators. 256 waves = 32 blocks.
// ---------------------------------------------------------------------------
__global__ void __launch_bounds__(256) iter_kernel(
    const float* __restrict__ A, const float* __restrict__ Xin,
    const float* __restrict__ BU, float* __restrict__ Xout) {
  const int lane = threadIdx.x & 31;
  const int w = threadIdx.x >> 5;
  const int half = lane >> 4;
  const int r = lane & 15;
  const int waveId = blockIdx.x * 8 + w;
  const int mTile = (waveId >> 2) * 16;
  const int nTile = (waveId & 3) * 64;

  v8f c0 = {}, c1 = {}, c2 = {}, c3 = {};
  for (int k0 = 0; k0 < 1024; k0 += 4) {
    v2f a = ldfragA(A, mTile + r, k0, 1024, half);
    v2f b0 = ldfragB_kn(Xin, nTile + r, k0, 256, half);
    v2f b1 = ldfragB_kn(Xin, nTile + r + 16, k0, 256, half);
    v2f b2 = ldfragB_kn(Xin, nTile + r + 32, k0, 256, half);
    v2f b3 = ldfragB_kn(Xin, nTile + r + 48, k0, 256, half);
    c0 = wmma_f32(a, b0, c0);
    c1 = wmma_f32(a, b1, c1);
    c2 = wmma_f32(a, b2, c2);
    c3 = wmma_f32(a, b3, c3);
  }
  const int col = nTile + r;
#pragma unroll
  for (int i = 0; i < 8; ++i) {
    const int row = mTile + half * 8 + i;
    const float* bu = BU + (size_t)row * 256 + col;
    float* o = Xout + (size_t)row * 256 + col;
    o[0] = fmaxf(c0[i] + bu[0], 0.0f);
    o[16] = fmaxf(c1[i] + bu[16], 0.0f);
    o[32] = fmaxf(c2[i] + bu[32], 0.0f);
    o[48] = fmaxf(c3[i] + bu[48], 0.0f);
  }
}

// ---------------------------------------------------------------------------
// Kernel 6: out = (C @ X + D @ U^T)^T  -> out is (m=256, q=256) row-major.
// GEMM dims: M=q=256, N=m=256; K1=1024 (C@X), K2=512 (D@Ut). 64 waves.
// ---------------------------------------------------------------------------
__global__ void __launch_bounds__(256) final_kernel(
    const float* __restrict__ Cm, const float* __restrict__ X,
    const float* __restrict__ D, const float* __restrict__ U,
    float* __restrict__ out) {
  const int lane = threadIdx.x & 31;
  const int w = threadIdx.x >> 5;
  const int half = lane >> 4;
  const int r = lane & 15;
  const int waveId = blockIdx.x * 8 + w;
  const int mTile = (waveId >> 2) * 16;  // q tiles (16)
  const int nTile = (waveId & 3) * 64;   // m tiles (4)

  v8f c0 = {}, c1 = {}, c2 = {}, c3 = {};
  for (int k0 = 0; k0 < 1024; k0 += 4) {  // C @ X
    v2f a = ldfragA(Cm, mTile + r, k0, 1024, half);
    v2f b0 = ldfragB_kn(X, nTile + r, k0, 256, half);
    v2f b1 = ldfragB_kn(X, nTile + r + 16, k0, 256, half);
    v2f b2 = ldfragB_kn(X, nTile + r + 32, k0, 256, half);
    v2f b3 = ldfragB_kn(X, nTile + r + 48, k0, 256, half);
    c0 = wmma_f32(a, b0, c0);
    c1 = wmma_f32(a, b1, c1);
    c2 = wmma_f32(a, b2, c2);
    c3 = wmma_f32(a, b3, c3);
  }
  for (int k0 = 0; k0 < 512; k0 += 4) {  // D @ U^T
    v2f a = ldfragA(D, mTile + r, k0, 512, half);
    v2f b0 = ldfragB_nk(U, nTile + r, k0, 512, half);
    v2f b1 = ldfragB_nk(U, nTile + r + 16, k0, 512, half);
    v2f b2 = ldfragB_nk(U, nTile + r + 32, k0, 512, half);
    v2f b3 = ldfragB_nk(U, nTile + r + 48, k0, 512, half);
    c0 = wmma_f32(a, b0, c0);
    c1 = wmma_f32(a, b1, c1);
    c2 = wmma_f32(a, b2, c2);
    c3 = wmma_f32(a, b3, c3);
  }
  // Transposed store: out[m][q] = Z[q][m]
  const int colm = nTile + r;
#pragma unroll
  for (int i = 0; i < 8; ++i) {
    const int rowq = mTile + half * 8 + i;
    out[(size_t)(colm)*256 + rowq] = c0[i];
    out[(size_t)(colm + 16) * 256 + rowq] = c1[i];
    out[(size_t)(colm + 32) * 256 + rowq] = c2[i];
    out[(size_t)(colm + 48) * 256 + rowq] = c3[i];
  }
}

// ---------------------------------------------------------------------------
extern "C" void kernel_launch(void* const* d_in, const int* in_sizes, int n_in,
                              void* d_out, int out_size, void* d_ws,
                              size_t ws_size, hipStream_t stream) {
  (void)in_sizes;
  (void)n_in;
  (void)out_size;
  (void)ws_size;
  const float* U = (const float*)d_in[0];  // (256, 512)
  const float* L = (const float*)d_in[1];  // (1024, 64)
  const float* R = (const float*)d_in[2];  // (1024, 64)
  const float* B = (const float*)d_in[3];  // (1024, 512)
  const float* C = (const float*)d_in[4];  // (256, 1024)
  const float* D = (const float*)d_in[5];  // (256, 512)
  float* out = (float*)d_out;              // (256, 256)

  // Workspace layout (floats)
  float* wsf = (float*)d_ws;
  float* scal = wsf;                          // [2] norms
  float* Amat = wsf + 16;                     // 1024*1024
  float* BU = Amat + 1024 * 1024;             // 1024*256
  float* X0 = BU + 1024 * 256;                // 1024*256
  float* X1 = X0 + 1024 * 256;                // 1024*256

  norms_kernel<<<dim3(2), dim3(1024), 0, stream>>>(L, R, scal);
  amat_kernel<<<dim3(128), dim3(256), 0, stream>>>(L, R, scal, Amat);
  bu_kernel<<<dim3(32), dim3(256), 0, stream>>>(B, U, BU);

  // Iteration 1: X = relu(A*0 + BU) = relu(BU)
  relu_copy_kernel<<<dim3(1024), dim3(256), 0, stream>>>(BU, X0, 1024 * 256);

  // Iterations 2..150
  float* cur = X0;
  float* nxt = X1;
  for (int it = 0; it < 149; ++it) {
    iter_kernel<<<dim3(32), dim3(256), 0, stream>>>(Amat, cur, BU, nxt);
    float* t = cur;
    cur = nxt;
    nxt = t;
  }

  final_kernel<<<dim3(8), dim3(256), 0, stream>>>(C, cur, D, U, out);
}